// YOLOv3Loss_75960791597734
// MI455X (gfx1250) — compile-verified
//
#include <hip/hip_runtime.h>

// YOLOv3 loss for MI455X (gfx1250, wave32).
// B=64, L=60, NC=10, D=15, grids {19,38,76}, LOSS_SCALE=1.

#define BATCH  64
#define LTAGS  60
#define DCH    15
#define NBLK1  512
#define TPB    256

typedef __attribute__((ext_vector_type(2))) float v2f;
typedef __attribute__((ext_vector_type(8))) float v8f;

__device__ __forceinline__ float bce_from_logit(float o, float t) {
    // matches reference: sigmoid, then clamped log
    float p  = 1.0f / (1.0f + expf(-o));
    float lp = fmaxf(logf(p),        -100.0f);
    float lq = fmaxf(logf(1.0f - p), -100.0f);
    return -(t * lp + (1.0f - t) * lq);
}

__device__ __forceinline__ float wave_sum(float v) {
    #pragma unroll
    for (int m = 16; m > 0; m >>= 1) v += __shfl_xor(v, m, 32);
    return v;
}

// ---------------------------------------------------------------------------
// Kernel 1: baseline confidence BCE over every cell (channel 4 only).
// One fixed partial per block -> deterministic two-stage reduction.
// ---------------------------------------------------------------------------
__global__ void conf_base_kernel(const float* __restrict__ out,
                                 const float* __restrict__ lab,
                                 float* __restrict__ partial, int HW) {
    const int tid = threadIdx.x;
    const int N = BATCH * 3 * HW;
    float acc = 0.0f;
    for (int i = blockIdx.x * TPB + tid; i < N; i += NBLK1 * TPB) {
        int hw = i % HW;
        int ba = i / HW;
        int a  = ba % 3;
        int b  = ba / 3;
        float o = out[(b * 45 + a * DCH + 4) * HW + hw];       // [B,45,H,W]
        float t = lab[(b * HW + hw) * 45 + a * DCH + 4];       // [B,H,W,3,15]
        acc += bce_from_logit(o, t);
    }
    __shared__ float sm[TPB / 32];
    float w = wave_sum(acc);
    if ((tid & 31) == 0) sm[tid >> 5] = w;
    __syncthreads();
    if (tid < 32) {
        float v = (tid < TPB / 32) ? sm[tid] : 0.0f;
        v = wave_sum(v);
        if (tid == 0) partial[blockIdx.x] = v;
    }
}

// ---------------------------------------------------------------------------
// Kernel 2: sparse corrections at tagged cells.
//   t==1 (ignore): subtract conf BCE at that cell (mask set to 0).
//   t==2 (pos):    add class BCE sum and lam * box-SSE.
// Dedup per category reproduces idempotent scatter-set semantics.
// ---------------------------------------------------------------------------
__global__ void tag_corr_kernel(const float* __restrict__ out_a, const float* __restrict__ lab_a, const int* __restrict__ tag_a,
                                const float* __restrict__ out_b, const float* __restrict__ lab_b, const int* __restrict__ tag_b,
                                const float* __restrict__ out_c, const float* __restrict__ lab_c, const int* __restrict__ tag_c,
                                const float* __restrict__ anchors, float* __restrict__ ws) {
    const int gtid      = blockIdx.x * TPB + threadIdx.x;
    const int per_scale = BATCH * LTAGS;
    int s   = gtid / per_scale;
    int rem = gtid - s * per_scale;
    int b   = rem / LTAGS;
    int l   = rem - b * LTAGS;

    const float* out; const float* lab; const int* tag; int H;
    if (s == 0)      { out = out_a; lab = lab_a; tag = tag_a; H = 19; }
    else if (s == 1) { out = out_b; lab = lab_b; tag = tag_b; H = 38; }
    else             { out = out_c; lab = lab_c; tag = tag_c; H = 76; }
    const int HW = H * H;

    float dconf = 0.0f, dcls = 0.0f, dpos = 0.0f;
    if (gtid < 3 * per_scale) {
        const int* trow = tag + (b * LTAGS) * 4;
        // valid = cumprod(tag0 != -1): prefix AND up to and including l
        bool valid = true;
        for (int j = 0; j <= l; ++j)
            if (trow[j * 4] == -1) { valid = false; break; }

        int t0 = trow[l * 4 + 0], t1 = trow[l * 4 + 1];
        int t2 = trow[l * 4 + 2], tt = trow[l * 4 + 3];
        int r  = min(max(t0, 0), H - 1);
        int c  = min(max(t1, 0), H - 1);
        int bx = min(max(t2, 0), 2);

        if (valid && (tt == 1 || tt == 2)) {
            bool dup = false;
            for (int j = 0; j < l; ++j) {            // all j<l are valid
                if (trow[j * 4 + 3] != tt) continue;
                int jr = min(max(trow[j * 4 + 0], 0), H - 1);
                int jc = min(max(trow[j * 4 + 1], 0), H - 1);
                int jb = min(max(trow[j * 4 + 2], 0), 2);
                if (jr == r && jc == c && jb == bx) { dup = true; break; }
            }
            if (!dup) {
                const int outBase = (b * 45 + bx * DCH) * HW + r * H + c;
                const int labBase = (b * HW + r * H + c) * 45 + bx * DCH;
                if (tt == 1) {
                    dconf = -bce_from_logit(out[outBase + 4 * HW], lab[labBase + 4]);
                } else {
                    #pragma unroll
                    for (int k = 5; k < DCH; ++k)
                        dcls += bce_from_logit(out[outBase + k * HW], lab[labBase + k]);
                    float p0 = 1.0f / (1.0f + expf(-out[outBase        ]));
                    float p1 = 1.0f / (1.0f + expf(-out[outBase + 1*HW ]));
                    float p2 = out[outBase + 2 * HW];
                    float p3 = out[outBase + 3 * HW];
                    float d0 = p0 - lab[labBase + 0], d1 = p1 - lab[labBase + 1];
                    float d2 = p2 - lab[labBase + 2], d3 = p3 - lab[labBase + 3];
                    float sq = d0 * d0 + d1 * d1 + d2 * d2 + d3 * d3;
                    float tw = lab[labBase + 2], th = lab[labBase + 3];
                    float a0 = anchors[(s * 3 + bx) * 2 + 0];
                    float a1 = anchors[(s * 3 + bx) * 2 + 1];
                    float lam = 2.0f - a0 * expf(tw) * (a1 * expf(th));
                    dpos = lam * sq;
                }
            }
        }
    }

    __shared__ float sm[3][TPB / 32];
    float wc = wave_sum(dconf), wl = wave_sum(dcls), wp = wave_sum(dpos);
    const int tid = threadIdx.x;
    if ((tid & 31) == 0) { sm[0][tid >> 5] = wc; sm[1][tid >> 5] = wl; sm[2][tid >> 5] = wp; }
    __syncthreads();
    if (tid < 32) {
        float vc = (tid < TPB / 32) ? sm[0][tid] : 0.0f;
        float vl = (tid < TPB / 32) ? sm[1][tid] : 0.0f;
        float vp = (tid < TPB / 32) ? sm[2][tid] : 0.0f;
        vc = wave_sum(vc); vl = wave_sum(vl); vp = wave_sum(vp);
        if (tid == 0) {
            ws[1536 + blockIdx.x] = vc;   // conf deltas join conf partial region
            ws[1600 + blockIdx.x] = vl;
            ws[1664 + blockIdx.x] = vp;
        }
    }
}

__global__ void zero_ws_kernel(float* __restrict__ ws) {
    int i = threadIdx.x;
    if (i < 192) ws[1536 + i] = 0.0f;     // zero padded partial slots [1536,1728)
}

// ---------------------------------------------------------------------------
// Kernel 3: deterministic final reduction via V_WMMA_F32_16X16X4_F32.
// A = ones(16x4); B carries 64 partials per step; C accumulates column sums
// (every C row identical), so total = sum of row 0 = lanes 0..15 of c[0].
// Launched as exactly one wave (EXEC all-ones, uniform control flow).
// ---------------------------------------------------------------------------
__global__ void finalize_kernel(const float* __restrict__ ws, float* __restrict__ outp) {
    const int lane = threadIdx.x;           // 0..31, one full wave
    v2f a; a.x = 1.0f; a.y = 1.0f;          // 16x4 all-ones A matrix
    const int bases[3] = {0, 1600, 1664};   // conf (1536 base + 64 deltas), cls, pos
    const int iters[3] = {25, 1, 1};        // 25*64 = 1600 conf partials
    float sums[3];
    for (int q = 0; q < 3; ++q) {
        v8f cacc = {};
        const float* p = ws + bases[q];
        for (int it = 0; it < iters[q]; ++it) {
            v2f bm;
            bm.x = p[it * 64 + lane];
            bm.y = p[it * 64 + 32 + lane];
            cacc = __builtin_amdgcn_wmma_f32_16x16x4_f32(
                false, a, false, bm, (short)0, cacc, false, false);
        }
        float v = cacc[0];                  // row of C; N = lane (0..15 per half)
        #pragma unroll
        for (int m = 8; m > 0; m >>= 1) v += __shfl_xor(v, m, 16);
        sums[q] = v;                        // lane 0: sum over N=0..15 of row 0
    }
    if (lane == 0) {
        const float inv = 1.0f / (float)BATCH;   // LOSS_SCALE == 1
        float conf = sums[0] * inv;
        float cls  = sums[1] * inv;
        float pos  = sums[2] * inv;
        outp[0] = conf + cls + pos;
        outp[1] = conf;
        outp[2] = cls;
        outp[3] = pos;
    }
}

extern "C" void kernel_launch(void* const* d_in, const int* in_sizes, int n_in,
                              void* d_out, int out_size, void* d_ws, size_t ws_size,
                              hipStream_t stream) {
    // setup_inputs() dict order: out_a, lab_a, tag_a, out_b, lab_b, tag_b,
    //                           out_c, lab_c, tag_c, anchors
    const float* out_a = (const float*)d_in[0];
    const float* lab_a = (const float*)d_in[1];
    const int*   tag_a = (const int*)  d_in[2];
    const float* out_b = (const float*)d_in[3];
    const float* lab_b = (const float*)d_in[4];
    const int*   tag_b = (const int*)  d_in[5];
    const float* out_c = (const float*)d_in[6];
    const float* lab_c = (const float*)d_in[7];
    const int*   tag_c = (const int*)  d_in[8];
    const float* anch  = (const float*)d_in[9];
    float* ws   = (float*)d_ws;
    float* outp = (float*)d_out;

    // ws layout (floats): [0,1536) conf block partials (3 scales x 512 blocks),
    // [1536,1600) conf deltas, [1600,1664) cls partials, [1664,1728) pos partials.
    zero_ws_kernel<<<1, 192, 0, stream>>>(ws);
    conf_base_kernel<<<NBLK1, TPB, 0, stream>>>(out_a, lab_a, ws,         19 * 19);
    conf_base_kernel<<<NBLK1, TPB, 0, stream>>>(out_b, lab_b, ws + 512,   38 * 38);
    conf_base_kernel<<<NBLK1, TPB, 0, stream>>>(out_c, lab_c, ws + 1024,  76 * 76);
    tag_corr_kernel<<<(3 * BATCH * LTAGS) / TPB, TPB, 0, stream>>>(
        out_a, lab_a, tag_a, out_b, lab_b, tag_b, out_c, lab_c, tag_c, anch, ws);
    finalize_kernel<<<1, 32, 0, stream>>>(ws, outp);
}